// AngleFreqEnhance_42107859370790
// MI455X (gfx1250) — compile-verified
//
#include <hip/hip_runtime.h>
#include <math.h>

typedef __attribute__((ext_vector_type(2))) float v2f;
typedef __attribute__((ext_vector_type(8))) float v8f;

constexpr int kH = 128, kW = 128, kHW = kH * kW;
constexpr int kB = 4, kCin = 256, kCmid = 16, kBins = 180;
constexpr int kRowP = 129;                     // padded LDS row stride (complex)
constexpr float kPi = 3.14159265358979323846f;
constexpr float kBw = 0.26179938779914946f;    // radians(15)
constexpr float kHighR2 = 368.64f;             // (0.3*64)^2

#define USE_ASYNC_LDS 1

// ---------------------------------------------------------------- FFT helper
__device__ inline void fft128(float2* a, int stride, const float2* tw, bool inv) {
  // bit-reversal (7 bits)
  for (int i = 0; i < 128; ++i) {
    int r = (int)(__brev((unsigned)i) >> 25);
    if (r > i) {
      float2 t = a[i * stride];
      a[i * stride] = a[r * stride];
      a[r * stride] = t;
    }
  }
  for (int s = 1; s <= 7; ++s) {
    int m = 1 << s, half = m >> 1;
    int tstep = 128 >> s;
    for (int k = 0; k < 128; k += m) {
      for (int j = 0; j < half; ++j) {
        float2 w = tw[j * tstep];
        float wy = inv ? -w.y : w.y;
        float2 u = a[(k + j) * stride];
        float2 v = a[(k + j + half) * stride];
        float tx = v.x * w.x - v.y * wy;
        float ty = v.x * wy + v.y * w.x;
        a[(k + j) * stride]        = make_float2(u.x + tx, u.y + ty);
        a[(k + j + half) * stride] = make_float2(u.x - tx, u.y - ty);
      }
    }
  }
}

// ------------------------------------------------------------ zero workspace
__global__ void zero_kernel(float* p, int n) {
  int i = blockIdx.x * blockDim.x + threadIdx.x;
  if (i < n) p[i] = 0.f;
}

// ------------------------------------------------- proj_in: [16,256]x[256,HW]
// One wave computes a 16(m) x 16(pixel) tile, K-loop over 256 via f32 WMMA.
__global__ __launch_bounds__(128) void proj_in_kernel(
    const float* __restrict__ x, const float* __restrict__ w_in,
    float* __restrict__ xproj) {
  int wave = blockIdx.x * 4 + (threadIdx.x >> 5);
  int lane = threadIdx.x & 31;
  int half = lane >> 4;
  int l = lane & 15;
  int b = wave >> 10;                 // 1024 pixel-tiles per batch
  int pix0 = (wave & 1023) << 4;

  const float* xb = x + (size_t)b * kCin * kHW + pix0 + l;
  const float* wrow = w_in + l * kCin + 2 * half;   // A row M=l, K base per half
  v8f acc = {0.f, 0.f, 0.f, 0.f, 0.f, 0.f, 0.f, 0.f};
  for (int k = 0; k < kCin; k += 4) {
    v2f a;  a.x = wrow[k];  a.y = wrow[k + 1];
    v2f bm;
    bm.x = xb[(size_t)(k + 2 * half) * kHW];
    bm.y = xb[(size_t)(k + 2 * half + 1) * kHW];
    acc = __builtin_amdgcn_wmma_f32_16x16x4_f32(false, a, false, bm,
                                                (short)0, acc, false, false);
  }
  float* dst = xproj + (size_t)b * kCmid * kHW + pix0 + l;
#pragma unroll
  for (int r = 0; r < 8; ++r)
    dst[(size_t)(r + 8 * half) * kHW] = acc[r];
}

// -------------------------- forward FFT (rows+cols in LDS) + angle histogram
__global__ __launch_bounds__(128) void fft_fwd_hist_kernel(
    const float* __restrict__ xproj, float2* __restrict__ F,
    float* __restrict__ hist) {
  __shared__ float2 sd[kH * kRowP];
  __shared__ float2 tw[64];
  __shared__ float hs[kBins];
  int img = blockIdx.x;               // b*16 + m
  int b = img >> 4;
  int tid = threadIdx.x;

  for (int i = tid; i < kBins; i += 128) hs[i] = 0.f;
  if (tid < 64) {
    float ang = -2.f * kPi * (float)tid / 128.f;
    tw[tid] = make_float2(cosf(ang), sinf(ang));
  }
  const float* src = xproj + (size_t)img * kHW;
#if USE_ASYNC_LDS
  // Async global->LDS staging of the real parts (ASYNCcnt path); imag zeroed
  // with plain DS stores. Each lane supplies its own padded LDS destination.
  for (int idx = tid; idx < kHW; idx += 128) {
    int i = idx >> 7, j = idx & 127;
    float2* slot = &sd[i * kRowP + j];
    slot->y = 0.f;
    unsigned lds_off = (unsigned)(size_t)slot;       // low dword = LDS offset
    unsigned long long gaddr = (unsigned long long)(size_t)(src + idx);
    asm volatile("global_load_async_to_lds_b32 %0, %1, off"
                 :: "v"(lds_off), "v"(gaddr) : "memory");
  }
  asm volatile("s_wait_asynccnt 0x0" ::: "memory");
#else
  for (int idx = tid; idx < kHW; idx += 128) {
    int i = idx >> 7, j = idx & 127;
    sd[i * kRowP + j] = make_float2(src[idx], 0.f);
  }
#endif
  __syncthreads();
  fft128(&sd[tid * kRowP], 1, tw, false);     // row FFT (thread = row)
  __syncthreads();
  fft128(&sd[tid], kRowP, tw, false);         // col FFT (thread = col)
  __syncthreads();

  float2* dstF = F + (size_t)img * kHW;
  for (int idx = tid; idx < kHW; idx += 128) {
    int i = idx >> 7, j = idx & 127;
    float2 v = sd[i * kRowP + j];
    dstF[idx] = v;
    float dy = (i < 64) ? (float)i : (float)(i - 128);  // shifted-grid coords
    float dx = (j < 64) ? (float)j : (float)(j - 128);
    if (dy * dy + dx * dx > kHighR2) {
      float mag = sqrtf(v.x * v.x + v.y * v.y);
      float th = atan2f(dy, dx) + kPi;
      float tm = fmodf(th, kPi);
      int bin = (int)(tm * ((float)kBins / kPi));
      bin = bin < 0 ? 0 : (bin > kBins - 1 ? kBins - 1 : bin);
      atomicAdd(&hs[bin], mag * (1.f / 16.f));          // channel mean folded in
    }
  }
  __syncthreads();
  for (int i = tid; i < kBins; i += 128)
    atomicAdd(&hist[b * kBins + i], hs[i]);
}

// ----------------------------------------------- per-batch smoothing + peaks
__global__ void peaks_kernel(const float* __restrict__ hist,
                             float* __restrict__ peaks) {
  int b = threadIdx.x;
  if (b >= kB) return;
  float h[kBins], e[kBins];
  const float* hb = hist + b * kBins;
  for (int i = 0; i < kBins; ++i) h[i] = hb[i];
  float sum = 0.f;
  for (int i = 0; i < kBins; ++i) {
    float p = (i > 0) ? h[i - 1] : 0.f;
    float n = (i < kBins - 1) ? h[i + 1] : 0.f;
    e[i] = 0.25f * p + 0.5f * h[i] + 0.25f * n;
    sum += e[i];
  }
  float mean = sum / (float)kBins;
  int amax = 0; float vmax = e[0];
  for (int i = 1; i < kBins; ++i)
    if (e[i] > vmax) { vmax = e[i]; amax = i; }
  int i1 = -1, i2 = -1, npk = 0;
  float v1 = -__builtin_inff(), v2 = -__builtin_inff();
  for (int i = 0; i < kBins; ++i) {
    float prev = e[(i + kBins - 1) % kBins];   // circular, matches jnp.roll
    float next = e[(i + 1) % kBins];
    if (e[i] > mean && e[i] > prev && e[i] > next) {
      ++npk;
      if (e[i] > v1)      { v2 = v1; i2 = i1; v1 = e[i]; i1 = i; }
      else if (e[i] > v2) { v2 = e[i]; i2 = i; }
    }
  }
  int p0, p1;
  if (npk == 0)      { p0 = amax; p1 = amax; }
  else if (npk == 1) { p0 = i1;   p1 = i1;   }
  else               { p0 = i1;   p1 = i2;   }
  const float cs = kPi / (float)kBins;
  peaks[b * 2 + 0] = ((float)p0 + 0.5f) * cs;
  peaks[b * 2 + 1] = ((float)p1 + 0.5f) * cs;
}

// ---------------------------------- gain * F, inverse FFT, take real -> xenh
__global__ __launch_bounds__(128) void gain_ifft_kernel(
    const float2* __restrict__ F, const float* __restrict__ peaks,
    float* __restrict__ xenh) {
  __shared__ float2 sd[kH * kRowP];
  __shared__ float2 tw[64];
  int img = blockIdx.x;
  int b = img >> 4;
  int tid = threadIdx.x;
  if (tid < 64) {
    float ang = -2.f * kPi * (float)tid / 128.f;
    tw[tid] = make_float2(cosf(ang), sinf(ang));
  }
  float pa0 = peaks[b * 2 + 0];
  float pa1 = peaks[b * 2 + 1];
  const float2* src = F + (size_t)img * kHW;
  for (int idx = tid; idx < kHW; idx += 128) {
    int i = idx >> 7, j = idx & 127;
    float dy = (i < 64) ? (float)i : (float)(i - 128);
    float dx = (j < 64) ? (float)j : (float)(j - 128);
    bool high = dy * dy + dx * dx > kHighR2;
    float th = atan2f(dy, dx) + kPi;
    float d0 = fabsf(th - pa0); d0 = fminf(d0, kPi - d0);
    float d1 = fabsf(th - pa1); d1 = fminf(d1, kPi - d1);
    bool enh = high && ((d0 <= kBw) || (d1 <= kBw));
    float g = enh ? 1.2f : (high ? 0.8f : 1.0f);
    float2 v = src[idx];
    sd[i * kRowP + j] = make_float2(v.x * g, v.y * g);
  }
  __syncthreads();
  fft128(&sd[tid * kRowP], 1, tw, true);
  __syncthreads();
  fft128(&sd[tid], kRowP, tw, true);
  __syncthreads();
  float* dst = xenh + (size_t)img * kHW;
  const float s = 1.f / 16384.f;                 // combined ortho normalization
  for (int idx = tid; idx < kHW; idx += 128) {
    int i = idx >> 7, j = idx & 127;
    dst[idx] = sd[i * kRowP + j].x * s;
  }
}

// --------------------------- proj_out: [256,16]x[16,HW] + residual, via WMMA
__global__ __launch_bounds__(256) void proj_out_kernel(
    const float* __restrict__ xenh, const float* __restrict__ w_out,
    const float* __restrict__ x, float* __restrict__ out) {
  int wave = blockIdx.x * 8 + (threadIdx.x >> 5);
  int lane = threadIdx.x & 31;
  int half = lane >> 4;
  int l = lane & 15;
  int b = wave >> 14;                 // 16 c-tiles * 1024 pixel-tiles per batch
  int rem = wave & 16383;
  int c0 = (rem >> 10) << 4;
  int pix0 = (rem & 1023) << 4;

  const float* eb = xenh + (size_t)b * kCmid * kHW + pix0 + l;
  const float* wrow = w_out + (c0 + l) * kCmid + 2 * half;
  v8f acc = {0.f, 0.f, 0.f, 0.f, 0.f, 0.f, 0.f, 0.f};
#pragma unroll
  for (int m = 0; m < kCmid; m += 4) {
    v2f a;  a.x = wrow[m];  a.y = wrow[m + 1];
    v2f bm;
    bm.x = eb[(size_t)(m + 2 * half) * kHW];
    bm.y = eb[(size_t)(m + 2 * half + 1) * kHW];
    acc = __builtin_amdgcn_wmma_f32_16x16x4_f32(false, a, false, bm,
                                                (short)0, acc, false, false);
  }
  size_t base = (size_t)b * kCin * kHW + pix0 + l;
#pragma unroll
  for (int r = 0; r < 8; ++r) {
    size_t o = base + (size_t)(c0 + r + 8 * half) * kHW;
    out[o] = x[o] + acc[r];
  }
}

// --------------------------------------------------------------------- launch
extern "C" void kernel_launch(void* const* d_in, const int* in_sizes, int n_in,
                              void* d_out, int out_size, void* d_ws, size_t ws_size,
                              hipStream_t stream) {
  const float* x     = (const float*)d_in[0];   // [4,256,128,128]
  const float* w_in  = (const float*)d_in[1];   // [16,256]
  const float* w_out = (const float*)d_in[2];   // [256,16]
  float* out = (float*)d_out;
  float* ws = (float*)d_ws;

  const size_t nproj = (size_t)kB * kCmid * kHW;   // 1,048,576 floats
  float*  xproj = ws;                               // also reused as x_enh
  float2* F     = (float2*)(ws + nproj);            // 2,097,152 floats
  float*  hist  = ws + nproj * 3;                   // B*180
  float*  pks   = hist + kB * kBins;                // B*2

  int nz = kB * kBins + kB * 2;
  zero_kernel<<<(nz + 255) / 256, 256, 0, stream>>>(hist, nz);
  proj_in_kernel<<<1024, 128, 0, stream>>>(x, w_in, xproj);
  fft_fwd_hist_kernel<<<kB * kCmid, 128, 0, stream>>>(xproj, F, hist);
  peaks_kernel<<<1, 32, 0, stream>>>(hist, pks);
  gain_ifft_kernel<<<kB * kCmid, 128, 0, stream>>>(F, pks, xproj);
  proj_out_kernel<<<8192, 256, 0, stream>>>(xproj, w_out, x, out);
}